// SelfFeature_16690242913101
// MI455X (gfx1250) — compile-verified
//
#include <hip/hip_runtime.h>

#define N_NODES   16384
#define IN_DIM    512
#define EMBED_DIM 256

typedef __attribute__((ext_vector_type(2))) float v2f;
typedef __attribute__((ext_vector_type(4))) float v4f;
typedef __attribute__((ext_vector_type(8))) float v8f;

// ---------------------------------------------------------------------------
// Kernel 1: d[row] = 1/(rowsum(adj)+1) + 1
// 1 GiB streaming read -> the roofline floor (~46 us @ 23.3 TB/s).
// One 256-thread block per row; float4 non-temporal loads (single-use data,
// keep it out of L2 so x/W/out stay resident there for the GEMM).
// ---------------------------------------------------------------------------
__global__ __launch_bounds__(256) void rowsum_scale_kernel(
    const float* __restrict__ adj, float* __restrict__ dvec) {
  const int row = blockIdx.x;
  const v4f* rp = reinterpret_cast<const v4f*>(adj + (size_t)row * N_NODES);

  float s = 0.0f;
  const int nv = N_NODES / 4;  // 4096 float4 per row
  for (int i = threadIdx.x; i < nv; i += 256) {
    v4f v = __builtin_nontemporal_load(rp + i);
    s += (v.x + v.y) + (v.z + v.w);
  }

  // wave32 reduction
  #pragma unroll
  for (int off = 16; off > 0; off >>= 1)
    s += __shfl_down(s, off, 32);

  __shared__ float ws[8];
  if ((threadIdx.x & 31) == 0) ws[threadIdx.x >> 5] = s;
  __syncthreads();

  if (threadIdx.x == 0) {
    float t = 0.0f;
    #pragma unroll
    for (int w = 0; w < 8; ++w) t += ws[w];
    dvec[row] = 1.0f / (t + 1.0f) + 1.0f;
  }
}

// ---------------------------------------------------------------------------
// Kernel 2: out = relu(diag(d) * (x @ W^T)) via V_WMMA_F32_16X16X4_F32.
// One wave per 16x16 output tile, K=512 in steps of 4 (128 WMMAs/wave).
// A-lane load:  x[n0 + (lane&15)][k0 + (lane>>4)*2 .. +1]   (float2)
// B-lane load:  W[e0 + (lane&15)][k0 + (lane>>4)*2 .. +1]   (float2)
// d-scale is applied once to the accumulator (commutes with the K-sum).
// ---------------------------------------------------------------------------
__global__ __launch_bounds__(256) void gcn_wmma_f32_kernel(
    const float* __restrict__ x, const float* __restrict__ W,
    const float* __restrict__ dvec, float* __restrict__ out) {
  const int lane  = threadIdx.x & 31;
  const int wave  = threadIdx.x >> 5;           // 0..7
  const int nTile = blockIdx.x * 8 + wave;      // 0..1023
  const int eTile = blockIdx.y;                 // 0..15
  const int n0 = nTile * 16;
  const int e0 = eTile * 16;

  const int lhalf = lane >> 4;                  // 0 or 1
  const int lmod  = lane & 15;

  const float* ap = x + (size_t)(n0 + lmod) * IN_DIM + (lhalf << 1);
  const float* bp = W + (size_t)(e0 + lmod) * IN_DIM + (lhalf << 1);

  v8f acc = {};
  #pragma unroll 8
  for (int k0 = 0; k0 < IN_DIM; k0 += 4) {
    v2f a = *reinterpret_cast<const v2f*>(ap + k0);
    v2f b = *reinterpret_cast<const v2f*>(bp + k0);
    // (neg_a, A, neg_b, B, c_mod, C, reuse_a, reuse_b)
    acc = __builtin_amdgcn_wmma_f32_16x16x4_f32(
        false, a, false, b, (short)0, acc, false, false);
  }

  // C/D layout: VGPR r -> M = r + lhalf*8, N = lmod
  const int mbase = n0 + (lhalf << 3);
  v4f d0 = *reinterpret_cast<const v4f*>(dvec + mbase);
  v4f d1 = *reinterpret_cast<const v4f*>(dvec + mbase + 4);
  float ds[8] = {d0.x, d0.y, d0.z, d0.w, d1.x, d1.y, d1.z, d1.w};

  #pragma unroll
  for (int r = 0; r < 8; ++r) {
    float v = acc[r] * ds[r];
    out[(size_t)(mbase + r) * EMBED_DIM + (e0 + lmod)] = fmaxf(v, 0.0f);
  }
}

extern "C" void kernel_launch(void* const* d_in, const int* in_sizes, int n_in,
                              void* d_out, int out_size, void* d_ws, size_t ws_size,
                              hipStream_t stream) {
  const float* adj = (const float*)d_in[0];  // [N, N]
  const float* x   = (const float*)d_in[1];  // [N, IN_DIM]
  const float* W   = (const float*)d_in[2];  // [EMBED_DIM, IN_DIM]
  float* out  = (float*)d_out;               // [N, EMBED_DIM]
  float* dvec = (float*)d_ws;                // N floats of scratch (64 KB)

  rowsum_scale_kernel<<<N_NODES, 256, 0, stream>>>(adj, dvec);

  dim3 block(256);
  dim3 grid(N_NODES / (16 * 8), EMBED_DIM / 16);  // (128, 16)
  gcn_wmma_f32_kernel<<<grid, block, 0, stream>>>(x, W, dvec, out);
}